// HippoSsmLayer_16896401343117
// MI455X (gfx1250) — compile-verified
//
#include <hip/hip_runtime.h>
#include <hip/hip_bf16.h>

typedef float v2f __attribute__((ext_vector_type(2)));
typedef float v4f __attribute__((ext_vector_type(4)));
typedef float v8f __attribute__((ext_vector_type(8)));

#define SDIM 64
#define NCH  128
#define NB   8
#define SLEN 4096

// ---------------------------------------------------------------------------
// Setup kernel: bilinear discretization.
//   L = I - (dt/2) * A      (lower triangular, since HiPPO A is lower tri)
//   Linv = L^{-1}           (forward substitution, exact)
//   A_bar = 2*Linv - I      (also lower triangular!)
//   B_bar = dt * Linv @ B
// One block, 256 threads. Trivial cost.
// ---------------------------------------------------------------------------
__global__ void hippo_setup_kernel(const float* __restrict__ A,
                                   const float* __restrict__ Bm,
                                   const float* __restrict__ dtp,
                                   float* __restrict__ abar,
                                   float* __restrict__ bbar) {
    __shared__ float Lsh[SDIM * SDIM];
    __shared__ float Linv[SDIM * SDIM];
    const float dt = dtp[0];
    const int tid = threadIdx.x;

    for (int e = tid; e < SDIM * SDIM; e += 256) {
        const int i = e >> 6, j = e & 63;
        float l = -0.5f * dt * A[e];
        if (i == j) l += 1.0f;
        Lsh[e] = l;
    }
    __syncthreads();

    if (tid < SDIM) {
        const int j = tid;  // thread j solves column j of Linv
        for (int i = 0; i < SDIM; ++i) {
            if (i < j) {
                Linv[i * SDIM + j] = 0.0f;
            } else {
                float sum = (i == j) ? 1.0f : 0.0f;
                for (int k = j; k < i; ++k)
                    sum -= Lsh[i * SDIM + k] * Linv[k * SDIM + j];
                Linv[i * SDIM + j] = sum / Lsh[i * SDIM + i];
            }
        }
    }
    __syncthreads();

    for (int e = tid; e < SDIM * SDIM; e += 256) {
        const int i = e >> 6, j = e & 63;
        abar[e] = 2.0f * Linv[e] - ((i == j) ? 1.0f : 0.0f);
    }
    for (int e = tid; e < SDIM * NCH; e += 256) {
        const int i = e >> 7, c = e & 127;
        float sum = 0.0f;
        for (int k = 0; k <= i; ++k)
            sum += Linv[i * SDIM + k] * Bm[k * NCH + c];
        bbar[e] = dt * sum;
    }
}

// Tile (tn, tk) of A_bar is nonzero only in the lower triangle:
// rows n in [tn*16, tn*16+16), cols m in [tk*4, tk*4+4) -> zero if tk >= 4*(tn+1).
#define ABAR_TILE_NONZERO(tn, tk) ((tk) < 4 * ((tn) + 1))

// ---------------------------------------------------------------------------
// Scan kernel: one wave owns 16 (b,c)-columns; state S[64,16] lives in 4 WMMA
// f32 accumulators; A_bar lives in 40 nonzero lower-triangular A-fragments
// (v2f each, 80 VGPRs). Per timestep:
//   bv   <- relayout(S)          (32x ds_bpermute + v_cndmask, DS pipe
//                                 overlaps the WMMA/XDL pipe)
//   acc  <- B_bar * x            (bias as initial C operand)
//   acc  += A_bar @ S            (40 chained v_wmma_f32_16x16x4_f32,
//                                 chains of depth 4/8/12/16)
//   y    <- dot(C, S) + half-wave reduce
// Time is chunked by 4: one b128 load of x (double-buffered, prefetched under
// the previous chunk's WMMA work) and one b128 store of y per 4 steps.
// Grid: 8 blocks (one per batch) x 256 threads (8 waves covering 128 chans).
// ---------------------------------------------------------------------------
__global__ __launch_bounds__(256, 1)
void hippo_scan_kernel(const float* __restrict__ sig,
                       const float* __restrict__ Cmat,
                       const float* __restrict__ abar,
                       const float* __restrict__ bbar,
                       float* __restrict__ out) {
    const int b    = blockIdx.x;
    const int lane = threadIdx.x & 31;
    const int w    = threadIdx.x >> 5;
    const int lc   = lane & 15;        // column within tile / row within tile
    const int lh   = lane >> 4;        // half-wave
    const int c    = w * 16 + lc;      // channel

    // A_bar fragments (lower-triangular tiles only).
    // A-layout: lane holds M=lc, VGPR v holds K = tk*4 + v + 2*lh.
    v2f a[4][16];
#pragma unroll
    for (int tn = 0; tn < 4; ++tn)
#pragma unroll
        for (int tk = 0; tk < 16; ++tk) {
            if (ABAR_TILE_NONZERO(tn, tk)) {
                const int base = (tn * 16 + lc) * SDIM + tk * 4 + 2 * lh;
                a[tn][tk] = *reinterpret_cast<const v2f*>(abar + base);  // 8B aligned
            }
        }

    // B_bar and C weights in C/D accumulator layout: VGPR r holds n = tn*16+r+8*lh.
    float bb[4][8], cw[4][8];
#pragma unroll
    for (int tn = 0; tn < 4; ++tn)
#pragma unroll
        for (int r = 0; r < 8; ++r) {
            const int n = tn * 16 + r + 8 * lh;
            bb[tn][r] = bbar[n * NCH + c];
            cw[tn][r] = Cmat[n * NCH + c];
        }

    // State accumulators, zero-initialized.
    v8f s[4];
#pragma unroll
    for (int tn = 0; tn < 4; ++tn)
#pragma unroll
        for (int r = 0; r < 8; ++r) s[tn][r] = 0.0f;

    const int colOff = (b * NCH + c) * SLEN;
    const float* xp = sig + colOff;
    float*       yp = out + colOff;

    // Double-buffered 4-wide input chunk (16B aligned: colOff % 4096 == 0).
    v4f xcur4 = *reinterpret_cast<const v4f*>(xp);

#pragma unroll 1
    for (int tc = 0; tc < SLEN; tc += 4) {
        // Prefetch next chunk of x under this chunk's WMMA work.
        v4f xnext4;
        if (tc + 4 < SLEN) {
            xnext4 = *reinterpret_cast<const v4f*>(xp + tc + 4);
        } else {
            xnext4[0] = 0.0f; xnext4[1] = 0.0f; xnext4[2] = 0.0f; xnext4[3] = 0.0f;
        }

        v4f y4;
#pragma unroll
        for (int u = 0; u < 4; ++u) {
            const float xcur = xcur4[u];

            // --- Relayout state (C/D layout) into B operands (K = v + 2*lh) ---
            v2f bv[16];
#pragma unroll
            for (int tk = 0; tk < 16; ++tk) {
                float vals[2];
#pragma unroll
                for (int v = 0; v < 2; ++v) {
                    const int Rlow   = 4 * tk + v;   // row wanted by lanes 0-15
                    const int tn     = Rlow >> 4;
                    const int rr     = Rlow & 15;
                    const int half   = rr >> 3;      // half-wave holding these rows
                    const int regLo  = rr & 7;
                    const int regHi  = regLo + 2;    // row wanted by lanes 16-31
                    const float lowV  = s[tn][regLo];
                    const float highV = s[tn][regHi];
                    const float sw = __shfl_xor(half ? lowV : highV, 16, 32);
                    const float direct = half ? highV : lowV;
                    vals[v] = (lh == half) ? direct : sw;
                }
                bv[tk].x = vals[0];
                bv[tk].y = vals[1];
            }

            // --- S_new = A_bar @ S + B_bar * x  (bias as initial C operand;
            //     only lower-triangular tiles of A_bar contribute) ---
#pragma unroll
            for (int tn = 0; tn < 4; ++tn) {
                v8f acc;
#pragma unroll
                for (int r = 0; r < 8; ++r) acc[r] = bb[tn][r] * xcur;
#pragma unroll
                for (int tk = 0; tk < 16; ++tk) {
                    if (ABAR_TILE_NONZERO(tn, tk)) {
                        acc = __builtin_amdgcn_wmma_f32_16x16x4_f32(
                            /*neg_a=*/false, a[tn][tk],
                            /*neg_b=*/false, bv[tk],
                            /*c_mod=*/(short)0, acc,
                            /*reuse_a=*/false, /*reuse_b=*/false);
                    }
                }
                s[tn] = acc;  // bv already captured old state; safe to overwrite
            }

            // --- y = sum_n C[n,c] * S[n,col] ---
            float y = 0.0f;
#pragma unroll
            for (int tn = 0; tn < 4; ++tn)
#pragma unroll
                for (int r = 0; r < 8; ++r) y += cw[tn][r] * s[tn][r];
            y += __shfl_xor(y, 16, 32);  // combine the two half-wave row ranges
            y4[u] = y;
        }

        // One 16B store per 4 steps (lanes 0-15 carry the reduced values).
        if (lh == 0) *reinterpret_cast<v4f*>(yp + tc) = y4;

        xcur4 = xnext4;
    }
}

// ---------------------------------------------------------------------------
// Inputs (setup_inputs order): signal[8,128,4096] f32, B[64,128] f32,
// C[64,128] f32, A[64,64] f32, dt (scalar f32). Output: [8,128,4096] f32.
// ---------------------------------------------------------------------------
extern "C" void kernel_launch(void* const* d_in, const int* in_sizes, int n_in,
                              void* d_out, int out_size, void* d_ws, size_t ws_size,
                              hipStream_t stream) {
    const float* sig = (const float*)d_in[0];
    const float* Bm  = (const float*)d_in[1];
    const float* Cm  = (const float*)d_in[2];
    const float* Am  = (const float*)d_in[3];
    const float* dtp = (const float*)d_in[4];
    float* out  = (float*)d_out;
    float* abar = (float*)d_ws;                 // 64*64 floats
    float* bbar = abar + SDIM * SDIM;           // 64*128 floats

    hippo_setup_kernel<<<1, 256, 0, stream>>>(Am, Bm, dtp, abar, bbar);
    hippo_scan_kernel<<<NB, 256, 0, stream>>>(sig, Cm, abar, bbar, out);
}